// MambaBlock_54065048322123
// MI455X (gfx1250) — compile-verified
//
#include <hip/hip_runtime.h>
#include <math.h>

// ---------------- problem constants ----------------
#define D_MODEL 96
#define N_STATE 16
#define DI      192
#define RNK     6
#define KG      4
#define BATCH   4
#define Hh      64
#define Ww      64
#define L       (Hh*Ww)           // 4096
#define CDIM    (RNK + 2*N_STATE) // 38
#define CH      256               // scan chunk length
#define NSEG    (L/CH)            // 16 chunks
#define EPSV    1e-5f

typedef float v2f __attribute__((ext_vector_type(2)));
typedef float v8f __attribute__((ext_vector_type(8)));

__device__ __forceinline__ float fast_rcp(float x){ return __builtin_amdgcn_rcpf(x); }
__device__ __forceinline__ float silu_f(float x){ return x * fast_rcp(1.f + __expf(-x)); }
__device__ __forceinline__ float softplus_f(float x){ return x > 20.f ? x : log1pf(__expf(x)); }

// Position in row-major xc that scan direction k reads at step l.
__device__ __forceinline__ int dir_idx(int k, int l){
    int ll = (k & 2) ? (L - 1 - l) : l;
    if (k & 1){ int w_ = ll / Hh; int h_ = ll % Hh; return h_ * Ww + w_; }
    return ll;
}

// ---------------- WMMA f32 16x16x4 (codegen-confirmed round 1) -------------
__device__ __forceinline__ v8f wmma_f32(v2f a, v2f b, v8f c){
    // (neg_a, A, neg_b, B, c_mod, C, reuse_a, reuse_b)
    return __builtin_amdgcn_wmma_f32_16x16x4_f32(false, a, false, b, (short)0, c, false, false);
}

// ---------------- async global->LDS copies (CDNA5 ASYNCcnt path) -----------
#if __has_builtin(__builtin_amdgcn_global_load_async_to_lds_b32) && \
    __has_builtin(__builtin_amdgcn_s_wait_asynccnt)
#define HAVE_ASYNC_LDS 1
#define GLOBAL_AS __attribute__((address_space(1)))
#define LDS_AS    __attribute__((address_space(3)))
__device__ __forceinline__ void async_copy_b32(const float* g, float* l){
    // builtin takes non-const AS1 int* src, AS3 int* dst (per hipcc diagnostic);
    // generic->AS1 keeps the 64-bit VA; generic->AS3 keeps addr[31:0] (LDS offset)
    __builtin_amdgcn_global_load_async_to_lds_b32(
        (GLOBAL_AS int*)(uintptr_t)g,
        (LDS_AS int*)(uint32_t)(uintptr_t)l, 0, 0);
}
__device__ __forceinline__ void async_wait0(){ __builtin_amdgcn_s_wait_asynccnt(0); }
#else
#define HAVE_ASYNC_LDS 0
#endif

// ---------------- 1) input LayerNorm over C=96 ----------------
__global__ void k_ln_in(const float* __restrict__ x, const float* __restrict__ g,
                        const float* __restrict__ bt, float* __restrict__ xn){
    int idx = blockIdx.x * blockDim.x + threadIdx.x;   // b*L + l
    if (idx >= BATCH * L) return;
    int b = idx / L, l = idx % L;
    const float* xp = x + (size_t)b * D_MODEL * L + l; // NCHW input, stride L per channel
    float s = 0.f, ss = 0.f;
    for (int c = 0; c < D_MODEL; ++c){ float v = xp[(size_t)c * L]; s += v; ss += v * v; }
    float m = s / D_MODEL;
    float var = ss / D_MODEL - m * m;
    float r = rsqrtf(var + EPSV);
    float* o = xn + (size_t)idx * D_MODEL;             // NHWC out
    for (int c = 0; c < D_MODEL; ++c)
        o[c] = (xp[(size_t)c * L] - m) * r * g[c] + bt[c];
}

// ---------------- 2) in_proj GEMM: (16384 x 96) @ (96 x 384) via WMMA ------
// One wave: 1 M-tile x 4 N-tiles -> A fragment reused 4x per k-step.
__global__ void k_inproj(const float* __restrict__ xn, const float* __restrict__ wproj,
                         float* __restrict__ xq, float* __restrict__ z){
    const int wave = (blockIdx.x * blockDim.x + threadIdx.x) >> 5;
    const int lane = threadIdx.x & 31;
    const int NG = (2 * DI) / 64;                 // 6 groups of 4 col tiles
    const int mt = wave / NG, ng = wave % NG;
    const int mrow = mt * 16 + (lane & 15);
    const int ncol0 = ng * 64 + (lane & 15);
    const int khalf = (lane >> 4) ? 2 : 0;        // lanes 16-31 hold K=2,3
    v8f acc[4] = {};
    for (int kk = 0; kk < D_MODEL; kk += 4){
        v2f a;
        a.x = xn[(size_t)mrow * D_MODEL + kk + khalf];
        a.y = xn[(size_t)mrow * D_MODEL + kk + khalf + 1];
        #pragma unroll
        for (int q = 0; q < 4; ++q){
            int ncol = ncol0 + q * 16;
            v2f b;
            b.x = wproj[(size_t)ncol * D_MODEL + kk + khalf];  // B[k][n] = W[n][k]
            b.y = wproj[(size_t)ncol * D_MODEL + kk + khalf + 1];
            acc[q] = wmma_f32(a, b, acc[q]);
        }
    }
    #pragma unroll
    for (int q = 0; q < 4; ++q){
        #pragma unroll
        for (int r = 0; r < 8; ++r){
            int m = mt * 16 + r + ((lane >> 4) ? 8 : 0);
            int c = ng * 64 + q * 16 + (lane & 15);
            float v = acc[q][r];
            if (c < DI) xq[(size_t)m * DI + c] = v;
            else        z [(size_t)m * DI + (c - DI)] = silu_f(v);
        }
    }
}

// ---------------- 3) depthwise 3x3 conv + bias + silu (NHWC) ----------------
__global__ void k_conv(const float* __restrict__ xq, const float* __restrict__ cw,
                       const float* __restrict__ cb, float* __restrict__ xc){
    int idx = blockIdx.x * blockDim.x + threadIdx.x;  // (b*L + l)*DI + d
    int d = idx % DI; int t = idx / DI;
    int l = t % L;    int b = t / L;
    int h = l / Ww, w = l % Ww;
    float acc = cb[d];
    #pragma unroll
    for (int dh = -1; dh <= 1; ++dh)
        #pragma unroll
        for (int dw = -1; dw <= 1; ++dw){
            int hh = h + dh, ww = w + dw;
            if (hh >= 0 && hh < Hh && ww >= 0 && ww < Ww)
                acc = fmaf(xq[((size_t)b * L + hh * Ww + ww) * DI + d],
                           cw[d * 9 + (dh + 1) * 3 + (dw + 1)], acc);
        }
    xc[(size_t)idx] = silu_f(acc);
}

// ---------------- 4) x_dbl GEMM per (b,k): (4096 x 192) @ (192 x 38pad48) ---
// One wave: 1 M-tile x all 3 N-tiles; A rows gathered via dir_idx.
__global__ void k_xdbl(const float* __restrict__ xc, const float* __restrict__ xpw,
                       float* __restrict__ xdbl){
    const int wave = (blockIdx.x * blockDim.x + threadIdx.x) >> 5;
    const int lane = threadIdx.x & 31;
    const int MT = L / 16;
    int t = wave;
    const int mt = t % MT; t /= MT;
    const int k  = t % KG; const int b = t / KG;
    const int mrow = mt * 16 + (lane & 15);
    const int khalf = (lane >> 4) ? 2 : 0;
    const float* arow = xc + ((size_t)b * L + dir_idx(k, mrow)) * DI;  // contiguous 192
    bool nvalid[3]; const float* bcol[3];
    #pragma unroll
    for (int q = 0; q < 3; ++q){
        int ncol = q * 16 + (lane & 15);
        nvalid[q] = ncol < CDIM;
        bcol[q] = xpw + ((size_t)k * CDIM + (nvalid[q] ? ncol : 0)) * DI; // B[d][c]=xpw[k][c][d]
    }
    v8f acc[3] = {};
    for (int kk = 0; kk < DI; kk += 4){
        v2f a;
        a.x = arow[kk + khalf]; a.y = arow[kk + khalf + 1];
        #pragma unroll
        for (int q = 0; q < 3; ++q){
            v2f bb;
            bb.x = nvalid[q] ? bcol[q][kk + khalf]     : 0.f;
            bb.y = nvalid[q] ? bcol[q][kk + khalf + 1] : 0.f;
            acc[q] = wmma_f32(a, bb, acc[q]);
        }
    }
    #pragma unroll
    for (int q = 0; q < 3; ++q){
        #pragma unroll
        for (int r = 0; r < 8; ++r){
            int l = mt * 16 + r + ((lane >> 4) ? 8 : 0);
            int c = q * 16 + (lane & 15);
            if (c < CDIM)
                xdbl[((size_t)(b * KG + k) * CDIM + c) * L + l] = acc[q][r];
        }
    }
}

// ---------------- 5) scan pass A: per-chunk local (P = prod dA, h local) ----
__global__ void k_scanA(const float* __restrict__ xdbl, const float* __restrict__ xc,
                        const float* __restrict__ dtw, const float* __restrict__ dtb,
                        const float* __restrict__ alog,
                        float* __restrict__ chP, float* __restrict__ chH){
    int blk = blockIdx.x;
    int s = blk % NSEG; blk /= NSEG;
    int k = blk % KG;   int b = blk / KG;
    int d = threadIdx.x;                                   // 0..191
    __shared__ float sdt[64][RNK];
    __shared__ float sB[64][N_STATE];
    const int kd = k * DI + d;
    float A[N_STATE];
    #pragma unroll
    for (int n = 0; n < N_STATE; ++n) A[n] = -__expf(alog[kd * N_STATE + n]);
    float wr[RNK];
    #pragma unroll
    for (int r = 0; r < RNK; ++r) wr[r] = dtw[kd * RNK + r];
    const float bias = dtb[kd];
    float h[N_STATE], P[N_STATE];
    #pragma unroll
    for (int n = 0; n < N_STATE; ++n){ h[n] = 0.f; P[n] = 1.f; }
    const float* xd = xdbl + (size_t)(b * KG + k) * CDIM * L;
    for (int tile = 0; tile < CH / 64; ++tile){
        int l0 = s * CH + tile * 64;
        __syncthreads();
        for (int i = threadIdx.x; i < 64 * (RNK + N_STATE); i += blockDim.x){
            int j = i / (RNK + N_STATE), f = i % (RNK + N_STATE);
            float* dst = (f < RNK) ? &sdt[j][f] : &sB[j][f - RNK];
            const float* src = xd + (size_t)f * L + l0 + j; // rows 0..5 dt, 6..21 B
#if HAVE_ASYNC_LDS
            async_copy_b32(src, dst);
#else
            *dst = *src;
#endif
        }
#if HAVE_ASYNC_LDS
        async_wait0();
#endif
        __syncthreads();
        for (int j = 0; j < 64; ++j){
            int l = l0 + j;
            float dt = bias;
            #pragma unroll
            for (int r = 0; r < RNK; ++r) dt = fmaf(sdt[j][r], wr[r], dt);
            dt = softplus_f(dt);
            float u  = xc[((size_t)b * L + dir_idx(k, l)) * DI + d];
            float du = dt * u;
            #pragma unroll
            for (int n = 0; n < N_STATE; ++n){
                float dA = __expf(dt * A[n]);
                P[n] *= dA;
                h[n] = fmaf(dA, h[n], du * sB[j][n]);
            }
        }
    }
    size_t base = ((size_t)((b * KG + k) * DI + d) * NSEG + s) * N_STATE;
    #pragma unroll
    for (int n = 0; n < N_STATE; ++n){ chP[base + n] = P[n]; chH[base + n] = h[n]; }
}

// ---------------- 6) scan pass B: cross-chunk prefix -----------------------
__global__ void k_scanB(const float* __restrict__ chP, const float* __restrict__ chH,
                        float* __restrict__ hinit){
    int t = blockIdx.x * blockDim.x + threadIdx.x;         // (b*K+k)*DI + d
    if (t >= BATCH * KG * DI) return;
    float h[N_STATE];
    #pragma unroll
    for (int n = 0; n < N_STATE; ++n) h[n] = 0.f;
    for (int s = 0; s < NSEG; ++s){
        size_t base = ((size_t)t * NSEG + s) * N_STATE;
        #pragma unroll
        for (int n = 0; n < N_STATE; ++n){
            hinit[base + n] = h[n];
            h[n] = fmaf(chP[base + n], h[n], chH[base + n]);
        }
    }
}

// ---------------- 7) scan pass C: local rerun with h0, emit y --------------
__global__ void k_scanC(const float* __restrict__ xdbl, const float* __restrict__ xc,
                        const float* __restrict__ dtw, const float* __restrict__ dtb,
                        const float* __restrict__ alog, const float* __restrict__ Ds,
                        const float* __restrict__ hinit, float* __restrict__ y4){
    int blk = blockIdx.x;
    int s = blk % NSEG; blk /= NSEG;
    int k = blk % KG;   int b = blk / KG;
    int d = threadIdx.x;
    __shared__ float sdt[64][RNK];
    __shared__ float sB[64][N_STATE];
    __shared__ float sC[64][N_STATE];
    const int kd = k * DI + d;
    float A[N_STATE];
    #pragma unroll
    for (int n = 0; n < N_STATE; ++n) A[n] = -__expf(alog[kd * N_STATE + n]);
    float wr[RNK];
    #pragma unroll
    for (int r = 0; r < RNK; ++r) wr[r] = dtw[kd * RNK + r];
    const float bias = dtb[kd];
    const float Dd = Ds[kd];
    float h[N_STATE];
    size_t hb = ((size_t)((b * KG + k) * DI + d) * NSEG + s) * N_STATE;
    #pragma unroll
    for (int n = 0; n < N_STATE; ++n) h[n] = hinit[hb + n];
    const float* xd = xdbl + (size_t)(b * KG + k) * CDIM * L;
    float* yout = y4 + ((size_t)(b * KG + k) * L) * DI + d;
    for (int tile = 0; tile < CH / 64; ++tile){
        int l0 = s * CH + tile * 64;
        __syncthreads();
        for (int i = threadIdx.x; i < 64 * CDIM; i += blockDim.x){
            int j = i / CDIM, f = i % CDIM;
            float* dst;
            if (f < RNK)                dst = &sdt[j][f];
            else if (f < RNK + N_STATE) dst = &sB[j][f - RNK];
            else                        dst = &sC[j][f - RNK - N_STATE];
            const float* src = xd + (size_t)f * L + l0 + j;
#if HAVE_ASYNC_LDS
            async_copy_b32(src, dst);
#else
            *dst = *src;
#endif
        }
#if HAVE_ASYNC_LDS
        async_wait0();
#endif
        __syncthreads();
        for (int j = 0; j < 64; ++j){
            int l = l0 + j;
            float dt = bias;
            #pragma unroll
            for (int r = 0; r < RNK; ++r) dt = fmaf(sdt[j][r], wr[r], dt);
            dt = softplus_f(dt);
            float u  = xc[((size_t)b * L + dir_idx(k, l)) * DI + d];
            float du = dt * u;
            float y = 0.f;
            #pragma unroll
            for (int n = 0; n < N_STATE; ++n){
                float dA = __expf(dt * A[n]);
                h[n] = fmaf(dA, h[n], du * sB[j][n]);
                y = fmaf(h[n], sC[j][n], y);
            }
            yout[(size_t)l * DI] = y + Dd * u;
        }
    }
}

// ---------------- 8) merge 4 directions + out LN + gate z ------------------
__global__ void k_combine(const float* __restrict__ y4, const float* __restrict__ z,
                          const float* __restrict__ g, const float* __restrict__ bt,
                          float* __restrict__ yz){
    int idx = blockIdx.x * blockDim.x + threadIdx.x;       // b*L + l
    if (idx >= BATCH * L) return;
    int b = idx / L, l = idx % L;
    int h = l / Ww, w = l % Ww;
    int tl = w * Hh + h;                                    // transpose map
    const float* y0 = y4 + (((size_t)(b * KG + 0) * L) + l)           * DI;
    const float* y1 = y4 + (((size_t)(b * KG + 1) * L) + tl)          * DI;
    const float* y2 = y4 + (((size_t)(b * KG + 2) * L) + (L - 1 - l)) * DI;
    const float* y3 = y4 + (((size_t)(b * KG + 3) * L) + (L - 1 - tl))* DI;
    float s = 0.f, ss = 0.f;
    for (int d = 0; d < DI; ++d){
        float v = y0[d] + y1[d] + y2[d] + y3[d];
        s += v; ss += v * v;
    }
    float m = s / DI;
    float var = ss / DI - m * m;
    float r = rsqrtf(var + EPSV);
    const float* zp = z + (size_t)idx * DI;
    float* o = yz + (size_t)idx * DI;
    for (int d = 0; d < DI; ++d){
        float v = y0[d] + y1[d] + y2[d] + y3[d];
        o[d] = ((v - m) * r * g[d] + bt[d]) * zp[d];
    }
}

// ---------------- 9) out_proj GEMM: (16384 x 192) @ (192 x 96) -------------
// One wave: 1 M-tile x 3 N-tiles (2 groups cover N=96).
__global__ void k_outproj(const float* __restrict__ yz, const float* __restrict__ wout,
                          float* __restrict__ out){
    const int wave = (blockIdx.x * blockDim.x + threadIdx.x) >> 5;
    const int lane = threadIdx.x & 31;
    const int NG = 2;                             // 2 groups of 3 tiles
    const int mt = wave / NG, ng = wave % NG;
    const int mrow = mt * 16 + (lane & 15);
    const int ncol0 = ng * 48 + (lane & 15);
    const int khalf = (lane >> 4) ? 2 : 0;
    v8f acc[3] = {};
    for (int kk = 0; kk < DI; kk += 4){
        v2f a;
        a.x = yz[(size_t)mrow * DI + kk + khalf];
        a.y = yz[(size_t)mrow * DI + kk + khalf + 1];
        #pragma unroll
        for (int q = 0; q < 3; ++q){
            int ncol = ncol0 + q * 16;
            v2f b;
            b.x = wout[(size_t)ncol * DI + kk + khalf];
            b.y = wout[(size_t)ncol * DI + kk + khalf + 1];
            acc[q] = wmma_f32(a, b, acc[q]);
        }
    }
    #pragma unroll
    for (int q = 0; q < 3; ++q){
        #pragma unroll
        for (int r = 0; r < 8; ++r){
            int M = mt * 16 + r + ((lane >> 4) ? 8 : 0);
            int c = ng * 48 + q * 16 + (lane & 15);
            int b = M / L, l = M % L;
            out[((size_t)b * D_MODEL + c) * L + l] = acc[q][r];   // NCHW output
        }
    }
}

// ---------------- launch ----------------
extern "C" void kernel_launch(void* const* d_in, const int* in_sizes, int n_in,
                              void* d_out, int out_size, void* d_ws, size_t ws_size,
                              hipStream_t stream) {
    const float* x     = (const float*)d_in[0];
    const float* g_in  = (const float*)d_in[1];
    const float* b_in  = (const float*)d_in[2];
    const float* wproj = (const float*)d_in[3];
    const float* cw    = (const float*)d_in[4];
    const float* cb    = (const float*)d_in[5];
    const float* xpw   = (const float*)d_in[6];
    const float* dtw   = (const float*)d_in[7];
    const float* dtb   = (const float*)d_in[8];
    const float* alog  = (const float*)d_in[9];
    const float* Ds    = (const float*)d_in[10];
    const float* g_out = (const float*)d_in[11];
    const float* b_out = (const float*)d_in[12];
    const float* wout  = (const float*)d_in[13];
    float* out = (float*)d_out;

    float* ws = (float*)d_ws;
    float* xn   = ws; ws += (size_t)BATCH * L * D_MODEL;
    float* xq   = ws; ws += (size_t)BATCH * L * DI;
    float* z    = ws; ws += (size_t)BATCH * L * DI;
    float* xc   = ws; ws += (size_t)BATCH * L * DI;
    float* xdbl = ws; ws += (size_t)BATCH * KG * CDIM * L;
    float* chP  = ws; ws += (size_t)BATCH * KG * DI * NSEG * N_STATE;
    float* chH  = ws; ws += (size_t)BATCH * KG * DI * NSEG * N_STATE;
    float* hini = ws; ws += (size_t)BATCH * KG * DI * NSEG * N_STATE;
    float* y4   = ws; ws += (size_t)BATCH * KG * L * DI;
    float* yz   = xq;  // xq is dead after conv; reuse for gated output

    k_ln_in  <<<(BATCH * L + 255) / 256, 256, 0, stream>>>(x, g_in, b_in, xn);
    k_inproj <<<((L / 16) * BATCH * 6) / 8, 256, 0, stream>>>(xn, wproj, xq, z);
    k_conv   <<<(BATCH * L * DI) / 256, 256, 0, stream>>>(xq, cw, cb, xc);
    k_xdbl   <<<(BATCH * KG * (L / 16)) / 8, 256, 0, stream>>>(xc, xpw, xdbl);
    k_scanA  <<<BATCH * KG * NSEG, DI, 0, stream>>>(xdbl, xc, dtw, dtb, alog, chP, chH);
    k_scanB  <<<(BATCH * KG * DI + 255) / 256, 256, 0, stream>>>(chP, chH, hini);
    k_scanC  <<<BATCH * KG * NSEG, DI, 0, stream>>>(xdbl, xc, dtw, dtb, alog, Ds, hini, y4);
    k_combine<<<(BATCH * L + 255) / 256, 256, 0, stream>>>(y4, z, g_out, b_out, yz);
    k_outproj<<<((L / 16) * BATCH * 2) / 8, 256, 0, stream>>>(yz, wout, out);
}